// GNNModel_10711648436497
// MI455X (gfx1250) — compile-verified
//
#include <hip/hip_runtime.h>

// ---------------------------------------------------------------------------
// CDNA5 (gfx1250) hetero-SAGE GNN. wave32. All dense transforms via
// V_WMMA_F32_16X16X4_F32 (fp32-exact). Edge aggregation via f32 atomics
// (destination arrays are L2-resident: 32MB / 6.4MB << 192MB L2).
// ---------------------------------------------------------------------------

typedef __attribute__((ext_vector_type(2))) float v2f;
typedef __attribute__((ext_vector_type(8))) float v8f;

__device__ __forceinline__ v8f wmma_k4(v2f a, v2f b, v8f c) {
  // D = A(16x4) * B(4x16) + C(16x16), fp32
  return __builtin_amdgcn_wmma_f32_16x16x4_f32(
      /*neg_a=*/false, a, /*neg_b=*/false, b,
      /*c_mod=*/(short)0, c, /*reuse_a=*/false, /*reuse_b=*/false);
}

// ------------------------- utility kernels ---------------------------------

__global__ void zero_kernel(float* __restrict__ p, long long n) {
  long long i = (long long)blockIdx.x * blockDim.x + threadIdx.x;
  if (i < n) p[i] = 0.0f;
}

__global__ void count_kernel(const int* __restrict__ dst, int E,
                             float* __restrict__ cnt) {
  int i = blockIdx.x * blockDim.x + threadIdx.x;
  if (i < E) atomicAdd(&cnt[dst[i]], 1.0f);
}

__global__ void addmat_kernel(const float* __restrict__ a,
                              const float* __restrict__ b,
                              float* __restrict__ o, int n) {
  int i = blockIdx.x * blockDim.x + threadIdx.x;
  if (i < n) o[i] = a[i] + b[i];
}

// Scatter 16 floats per edge: agg[dst] += Y[src]. 4 lanes per edge (float4).
__global__ void scatter16_kernel(const float* __restrict__ Y,
                                 const int* __restrict__ src,
                                 const int* __restrict__ dst, int E,
                                 float* __restrict__ agg) {
  long long i = (long long)blockIdx.x * blockDim.x + threadIdx.x;
  long long total = (long long)E * 4;
  if (i >= total) return;
  int e = (int)(i >> 2);
  int c = (int)(i & 3);
  int s = src[e];
  int d = dst[e];
  const float4 v = *reinterpret_cast<const float4*>(Y + (size_t)s * 16 + c * 4);
  float* p = agg + (size_t)d * 16 + c * 4;
  atomicAdd(p + 0, v.x);
  atomicAdd(p + 1, v.y);
  atomicAdd(p + 2, v.z);
  atomicAdd(p + 3, v.w);
}

// out = relu(root + aggA/max(cntA,1) + bA + aggB/max(cntB,1) + bB), in place.
__global__ void finalize_l1_kernel(float* __restrict__ XO,
                                   const float* __restrict__ aggA,
                                   const float* __restrict__ cntA,
                                   const float* __restrict__ bA,
                                   const float* __restrict__ aggB,
                                   const float* __restrict__ cntB,
                                   const float* __restrict__ bB, int M) {
  long long i = (long long)blockIdx.x * blockDim.x + threadIdx.x;
  if (i >= (long long)M * 16) return;
  int node = (int)(i >> 4);
  int f = (int)(i & 15);
  float ca = fmaxf(cntA[node], 1.0f);
  float cb = fmaxf(cntB[node], 1.0f);
  float v = XO[i] + aggA[i] / ca + bA[f] + aggB[i] / cb + bB[f];
  XO[i] = fmaxf(v, 0.0f);
}

// --------------------- layer-1 transform (K=64, 3x N=16) ------------------
// Y0 = X@W0, Y1 = X@W1, Y2 = X@W2 sharing one A-fragment stream.
// One wave per 16-row tile; 16 K-steps x 3 WMMAs = 48 v_wmma per tile.
__global__ __launch_bounds__(128) void l1_transform_kernel(
    const float* __restrict__ X, const float* __restrict__ W0,
    const float* __restrict__ W1, const float* __restrict__ W2,
    float* __restrict__ Y0, float* __restrict__ Y1, float* __restrict__ Y2,
    int M) {
  const int lane = threadIdx.x & 31;
  const int wave = threadIdx.x >> 5;
  const int wpb = blockDim.x >> 5;
  const int half = lane >> 4;   // 0: K pair {k,k+1}; 1: {k+2,k+3}
  const int mrow = lane & 15;   // row (A) / col (B,D)
  const int nTiles = (M + 15) >> 4;

  // Preload B fragments (B is 4x16 per step; col striped across lanes).
  v2f b0[16], b1[16], b2[16];
#pragma unroll
  for (int s = 0; s < 16; ++s) {
    int k = 4 * s + 2 * half;
    b0[s] = (v2f){W0[k * 16 + mrow], W0[(k + 1) * 16 + mrow]};
    b1[s] = (v2f){W1[k * 16 + mrow], W1[(k + 1) * 16 + mrow]};
    b2[s] = (v2f){W2[k * 16 + mrow], W2[(k + 1) * 16 + mrow]};
  }

  for (int tile = blockIdx.x * wpb + wave; tile < nTiles;
       tile += gridDim.x * wpb) {
    const int rbase = tile << 4;
    int arow = rbase + mrow;
    if (arow > M - 1) arow = M - 1;
    const float* xr = X + (size_t)arow * 64;
    v8f c0 = {}, c1 = {}, c2 = {};
#pragma unroll
    for (int s = 0; s < 16; ++s) {
      int k = 4 * s + 2 * half;
      v2f a = (v2f){xr[k], xr[k + 1]};
      c0 = wmma_k4(a, b0[s], c0);
      c1 = wmma_k4(a, b1[s], c1);
      c2 = wmma_k4(a, b2[s], c2);
    }
#pragma unroll
    for (int r = 0; r < 8; ++r) {
      int row = rbase + r + 8 * half;
      if (row < M) {
        size_t o = (size_t)row * 16 + mrow;
        Y0[o] = c0[r];
        Y1[o] = c1[r];
        Y2[o] = c2[r];
      }
    }
  }
}

// --------------- layer-2 fused (K=16, N=32, 3 accumulating GEMMs) ----------
// OUT = (A0/max(c0,1))@W0 + (A1/max(c1,1))@W1 + A2@W2 + bias0 + bias1
// Row-scale folded into the A-fragment load. 24 v_wmma per 16x32 tile.
__global__ __launch_bounds__(128) void l2_fused_kernel(
    const float* __restrict__ A0, const float* __restrict__ cnt0,
    const float* __restrict__ A1, const float* __restrict__ cnt1,
    const float* __restrict__ A2, const float* __restrict__ W0,
    const float* __restrict__ W1, const float* __restrict__ W2,
    const float* __restrict__ bias0, const float* __restrict__ bias1,
    float* __restrict__ OUT, int M) {
  const int lane = threadIdx.x & 31;
  const int wave = threadIdx.x >> 5;
  const int wpb = blockDim.x >> 5;
  const int half = lane >> 4;
  const int mrow = lane & 15;
  const int nTiles = (M + 15) >> 4;

  const float* Ws[3] = {W0, W1, W2};
  v2f b[3][4][2];  // [matrix][k-step][col-tile]
#pragma unroll
  for (int m = 0; m < 3; ++m)
#pragma unroll
    for (int s = 0; s < 4; ++s)
#pragma unroll
      for (int t = 0; t < 2; ++t) {
        int k = 4 * s + 2 * half;
        int col = 16 * t + mrow;
        b[m][s][t] = (v2f){Ws[m][k * 32 + col], Ws[m][(k + 1) * 32 + col]};
      }

  const float biasL = bias0[mrow] + bias1[mrow];
  const float biasH = bias0[16 + mrow] + bias1[16 + mrow];

  for (int tile = blockIdx.x * wpb + wave; tile < nTiles;
       tile += gridDim.x * wpb) {
    const int rbase = tile << 4;
    int arow = rbase + mrow;
    if (arow > M - 1) arow = M - 1;
    const float rs0 = 1.0f / fmaxf(cnt0[arow], 1.0f);
    const float rs1 = 1.0f / fmaxf(cnt1[arow], 1.0f);
    const float* a0p = A0 + (size_t)arow * 16;
    const float* a1p = A1 + (size_t)arow * 16;
    const float* a2p = A2 + (size_t)arow * 16;
    v8f cL = {}, cH = {};
#pragma unroll
    for (int s = 0; s < 4; ++s) {
      int k = 4 * s + 2 * half;
      v2f a0 = (v2f){a0p[k] * rs0, a0p[k + 1] * rs0};
      cL = wmma_k4(a0, b[0][s][0], cL);
      cH = wmma_k4(a0, b[0][s][1], cH);
      v2f a1 = (v2f){a1p[k] * rs1, a1p[k + 1] * rs1};
      cL = wmma_k4(a1, b[1][s][0], cL);
      cH = wmma_k4(a1, b[1][s][1], cH);
      v2f a2 = (v2f){a2p[k], a2p[k + 1]};
      cL = wmma_k4(a2, b[2][s][0], cL);
      cH = wmma_k4(a2, b[2][s][1], cH);
    }
#pragma unroll
    for (int r = 0; r < 8; ++r) {
      int row = rbase + r + 8 * half;
      if (row < M) {
        size_t o = (size_t)row * 32 + mrow;
        OUT[o] = cL[r] + biasL;
        OUT[o + 16] = cH[r] + biasH;
      }
    }
  }
}

// --------------------------------- host ------------------------------------

extern "C" void kernel_launch(void* const* d_in, const int* in_sizes, int n_in,
                              void* d_out, int out_size, void* d_ws,
                              size_t ws_size, hipStream_t stream) {
  const float* x_dev = (const float*)d_in[0];
  const float* x_task = (const float*)d_in[1];
  const int* wr_src = (const int*)d_in[2];
  const int* wr_dst = (const int*)d_in[3];
  const int* rv_src = (const int*)d_in[4];
  const int* rv_dst = (const int*)d_in[5];
  const int* wb_src = (const int*)d_in[6];
  const int* wb_dst = (const int*)d_in[7];
  const int* rb_src = (const int*)d_in[8];
  const int* rb_dst = (const int*)d_in[9];
  const float* p1wr_wl = (const float*)d_in[10];
  const float* p1wr_bl = (const float*)d_in[11];
  const float* p1wr_wr = (const float*)d_in[12];
  const float* p1rv_wl = (const float*)d_in[13];
  const float* p1rv_bl = (const float*)d_in[14];
  const float* p1rv_wr = (const float*)d_in[15];
  const float* p1wb_wl = (const float*)d_in[16];
  const float* p1wb_bl = (const float*)d_in[17];
  const float* p1wb_wr = (const float*)d_in[18];
  const float* p1rb_wl = (const float*)d_in[19];
  const float* p1rb_bl = (const float*)d_in[20];
  const float* p1rb_wr = (const float*)d_in[21];
  const float* p2wr_wl = (const float*)d_in[22];
  const float* p2wr_bl = (const float*)d_in[23];
  const float* p2wr_wr = (const float*)d_in[24];
  const float* p2rv_wl = (const float*)d_in[25];
  const float* p2rv_bl = (const float*)d_in[26];
  const float* p2rv_wr = (const float*)d_in[27];
  const float* p2wb_wl = (const float*)d_in[28];
  const float* p2wb_bl = (const float*)d_in[29];
  const float* p2wb_wr = (const float*)d_in[30];
  const float* p2rb_wl = (const float*)d_in[31];
  const float* p2rb_bl = (const float*)d_in[32];
  const float* p2rb_wr = (const float*)d_in[33];

  const int ND = in_sizes[0] / 64;  // dev nodes
  const int NT = in_sizes[1] / 64;  // task nodes
  const int E_wr = in_sizes[2], E_rv = in_sizes[4];
  const int E_wb = in_sizes[6], E_rb = in_sizes[8];

  // Workspace layout (floats). Counts + aggs are contiguous for one-shot zero.
  float* ws = (float*)d_ws;
  size_t off = 0;
  float* cnt_wr = ws + off; off += NT;
  float* cnt_rv = ws + off; off += NT;
  float* cnt_wb = ws + off; off += ND;
  float* cnt_rb = ws + off; off += ND;
  float* agg_wr = ws + off; off += (size_t)NT * 16;
  float* agg_rv = ws + off; off += (size_t)NT * 16;
  float* agg_wb = ws + off; off += (size_t)ND * 16;
  float* agg_rb = ws + off; off += (size_t)ND * 16;
  float* y_wr = ws + off; off += (size_t)ND * 16;
  float* y_rv = ws + off; off += (size_t)ND * 16;
  float* y_wb = ws + off; off += (size_t)NT * 16;
  float* y_rb = ws + off; off += (size_t)NT * 16;
  float* t1 = ws + off; off += (size_t)NT * 16;
  float* d1 = ws + off; off += (size_t)ND * 16;
  float* ws1t = ws + off; off += 64 * 16;  // p1wr_wr + p1rv_wr
  float* ws1d = ws + off; off += 64 * 16;  // p1wb_wr + p1rb_wr
  float* ws2t = ws + off; off += 16 * 32;  // p2wr_wr + p2rv_wr
  float* ws2d = ws + off; off += 16 * 32;  // p2wb_wr + p2rb_wr

  const int TPB = 256;
  const long long aggN = (long long)(2LL * NT + 2LL * ND) * 16;
  const long long zero1N = (long long)(2LL * NT + 2LL * ND) + aggN;

  // 1) zero counts + aggregators
  zero_kernel<<<(int)((zero1N + TPB - 1) / TPB), TPB, 0, stream>>>(ws, zero1N);

  // 2) neighbor counts (shared across both layers)
  count_kernel<<<(E_wr + TPB - 1) / TPB, TPB, 0, stream>>>(wr_dst, E_wr, cnt_wr);
  count_kernel<<<(E_rv + TPB - 1) / TPB, TPB, 0, stream>>>(rv_dst, E_rv, cnt_rv);
  count_kernel<<<(E_wb + TPB - 1) / TPB, TPB, 0, stream>>>(wb_dst, E_wb, cnt_wb);
  count_kernel<<<(E_rb + TPB - 1) / TPB, TPB, 0, stream>>>(rb_dst, E_rb, cnt_rb);

  // 3) summed root weights
  addmat_kernel<<<4, TPB, 0, stream>>>(p1wr_wr, p1rv_wr, ws1t, 64 * 16);
  addmat_kernel<<<4, TPB, 0, stream>>>(p1wb_wr, p1rb_wr, ws1d, 64 * 16);
  addmat_kernel<<<2, TPB, 0, stream>>>(p2wr_wr, p2rv_wr, ws2t, 16 * 32);
  addmat_kernel<<<2, TPB, 0, stream>>>(p2wb_wr, p2rb_wr, ws2d, 16 * 32);

  // 4) layer-1: transform sources first (64->16), so edges move 16 floats.
  //    Also produces root terms directly into t1 / d1.
  const int tilesD = (ND + 15) / 16, tilesT = (NT + 15) / 16;
  l1_transform_kernel<<<(tilesD + 3) / 4, 128, 0, stream>>>(
      x_dev, p1wr_wl, p1rv_wl, ws1d, y_wr, y_rv, d1, ND);
  l1_transform_kernel<<<(tilesT + 3) / 4, 128, 0, stream>>>(
      x_task, p1wb_wl, p1rb_wl, ws1t, y_wb, y_rb, t1, NT);

  // 5) layer-1 edge scatter (transformed space, 16 floats/edge)
  auto scb = [&](int E) { return (int)(((long long)E * 4 + TPB - 1) / TPB); };
  scatter16_kernel<<<scb(E_wr), TPB, 0, stream>>>(y_wr, wr_src, wr_dst, E_wr, agg_wr);
  scatter16_kernel<<<scb(E_rv), TPB, 0, stream>>>(y_rv, rv_src, rv_dst, E_rv, agg_rv);
  scatter16_kernel<<<scb(E_wb), TPB, 0, stream>>>(y_wb, wb_src, wb_dst, E_wb, agg_wb);
  scatter16_kernel<<<scb(E_rb), TPB, 0, stream>>>(y_rb, rb_src, rb_dst, E_rb, agg_rb);

  // 6) layer-1 finalize: mean + biases + root, ReLU (in place on t1/d1)
  finalize_l1_kernel<<<(int)(((long long)NT * 16 + TPB - 1) / TPB), TPB, 0,
                       stream>>>(t1, agg_wr, cnt_wr, p1wr_bl, agg_rv, cnt_rv,
                                 p1rv_bl, NT);
  finalize_l1_kernel<<<(int)(((long long)ND * 16 + TPB - 1) / TPB), TPB, 0,
                       stream>>>(d1, agg_wb, cnt_wb, p1wb_bl, agg_rb, cnt_rb,
                                 p1rb_bl, ND);

  // 7) re-zero aggregators for layer 2
  zero_kernel<<<(int)((aggN + TPB - 1) / TPB), TPB, 0, stream>>>(agg_wr, aggN);

  // 8) layer-2 edge scatter (aggregate-first: src dim is already 16)
  scatter16_kernel<<<scb(E_wr), TPB, 0, stream>>>(d1, wr_src, wr_dst, E_wr, agg_wr);
  scatter16_kernel<<<scb(E_rv), TPB, 0, stream>>>(d1, rv_src, rv_dst, E_rv, agg_rv);
  scatter16_kernel<<<scb(E_wb), TPB, 0, stream>>>(t1, wb_src, wb_dst, E_wb, agg_wb);
  scatter16_kernel<<<scb(E_rb), TPB, 0, stream>>>(t1, rb_src, rb_dst, E_rb, agg_rb);

  // 9) layer-2 fused GEMMs: mean-scale folded into A loads, bias fused.
  //    Output order: (d2, t2) concatenated flat.
  float* out_d2 = (float*)d_out;
  float* out_t2 = (float*)d_out + (size_t)ND * 32;
  l2_fused_kernel<<<(tilesT + 3) / 4, 128, 0, stream>>>(
      agg_wr, cnt_wr, agg_rv, cnt_rv, t1, p2wr_wl, p2rv_wl, ws2t, p2wr_bl,
      p2rv_bl, out_t2, NT);
  l2_fused_kernel<<<(tilesD + 3) / 4, 128, 0, stream>>>(
      agg_wb, cnt_wb, agg_rb, cnt_rb, d1, p2wb_wl, p2rb_wl, ws2d, p2wb_bl,
      p2rb_bl, out_d2, ND);
}